// Model_69853348102667
// MI455X (gfx1250) — compile-verified
//
#include <hip/hip_runtime.h>
#include <cmath>

// ---------------------------------------------------------------------------
// Hyena-style long-conv block for MI455X (gfx1250, wave32, WMMA, 320KB LDS).
// B=8, L=8192, D=256, proj->3D=768, MLP_HID=64, FFT size 2L=16384.
// ---------------------------------------------------------------------------

#define D_MODEL 256
#define SEQ_LEN 8192
#define NFFT    16384
#define MFFT    8192          // NFFT/2 butterflies per stage
#define BATCH   8
#define PROJ3   768
#define MLP_HID 64

typedef float        v2f __attribute__((ext_vector_type(2)));
typedef float        v8f __attribute__((ext_vector_type(8)));
typedef unsigned int v4u __attribute__((ext_vector_type(4)));
typedef int          v4i __attribute__((ext_vector_type(4)));
typedef int          v8i __attribute__((ext_vector_type(8)));

__device__ __forceinline__ float2 cmulc(float2 a, float2 b) {
    return make_float2(a.x * b.x - a.y * b.y, a.x * b.y + a.y * b.x);
}

// ---------------------------------------------------------------------------
// Stockham autosort radix-2 FFT, N=16384, entirely in LDS (ping-pong x<->y).
// 256 threads, 32 butterflies per thread per stage, 14 stages.
// dsign = -1 forward, +1 inverse (unscaled). Returns buffer holding result.
// Twiddle: w = exp(dsign * i*pi * (j*m)/8192) since l*m == 8192 at all stages.
// ---------------------------------------------------------------------------
__device__ float2* fft16384(float2* __restrict__ x, float2* __restrict__ y,
                            int tid, float dsign) {
    float2* src = x;
    float2* dst = y;
    int m = 1;
    #pragma unroll 1
    for (int stage = 0; stage < 14; ++stage) {
        #pragma unroll 4
        for (int it = 0; it < MFFT / 256; ++it) {
            int idx = tid + (it << 8);         // 0..8191
            int k   = idx & (m - 1);
            int jm  = idx - k;                 // j*m
            float ang = (float)jm * (1.0f / (float)MFFT);   // in [0,1)
            float s, c;
            sincospif(ang, &s, &c);
            float2 w  = make_float2(c, dsign * s);
            float2 c0 = src[idx];
            float2 c1 = src[idx + MFFT];
            float2 dif = make_float2(c0.x - c1.x, c0.y - c1.y);
            int d0 = (jm << 1) + k;
            dst[d0]     = make_float2(c0.x + c1.x, c0.y + c1.y);
            dst[d0 + m] = cmulc(w, dif);
        }
        __syncthreads();
        float2* tmp = src; src = dst; dst = tmp;
        m <<= 1;
    }
    return src;   // 14 stages (even) -> result back in x
}

// ---------------------------------------------------------------------------
// K0: implicit filter h[d][l] = W2[d]·silu(t*W1+b1) + b2[d], then forward FFT.
// One workgroup per d. Stores full complex spectrum Hf[d][0..16383].
// ---------------------------------------------------------------------------
__global__ void hfilter_fft_kernel(const float* __restrict__ t,
                                   const float* __restrict__ W1,
                                   const float* __restrict__ b1,
                                   const float* __restrict__ W2,
                                   const float* __restrict__ b2,
                                   float2* __restrict__ Hf) {
    extern __shared__ char smem[];
    float2* bufa = (float2*)smem;
    float2* bufb = bufa + NFFT;
    __shared__ float sW1[MLP_HID], sb1[MLP_HID], sW2[MLP_HID];
    const int tid = threadIdx.x;
    const int d   = blockIdx.x;
    if (tid < MLP_HID) {
        sW1[tid] = W1[tid];
        sb1[tid] = b1[tid];
        sW2[tid] = W2[d * MLP_HID + tid];
    }
    __syncthreads();
    const float bias = b2[d];
    for (int l = tid; l < NFFT; l += 256) {
        float val = 0.0f;
        if (l < SEQ_LEN) {
            float tt  = t[l];
            float acc = bias;
            #pragma unroll
            for (int j = 0; j < MLP_HID; ++j) {
                float z   = fmaf(tt, sW1[j], sb1[j]);
                float sig = 1.0f / (1.0f + __expf(-z));
                acc = fmaf(sW2[j], z * sig, acc);
            }
            val = acc;
        }
        bufa[l] = make_float2(val, 0.0f);
    }
    __syncthreads();
    float2* out = fft16384(bufa, bufb, tid, -1.0f);
    for (int k = tid; k < NFFT; k += 256)
        Hf[(size_t)d * NFFT + k] = out[k];
}

// ---------------------------------------------------------------------------
// GEMM  C[M,N] = A[M,256] * W[N,256]^T + bias[N]  using V_WMMA_F32_16X16X4_F32.
// Block tile: 128(M) x 64(N), 8 waves (each wave: 16 rows x 64 cols = 4 accs),
// K staged in LDS in chunks of 32.
//
// LDS tiles are m-major / n-major with k-stride 36 so that:
//  * each WMMA fragment (k, k+1) is one aligned 8-byte ds_load_b64
//    (m*36+k is even; gcd(36,64)=4 -> 16 fragment lanes hit distinct banks)
//  * each staging store of 4 consecutive k is one aligned ds_store_b128
//
// ATRANS=false : A row-major [M][256]           (input projection)
// ATRANS=true  : A element (m,k) at A[(batch*256+k)*8192 + l]  (output proj
//                reading ygate stored as [B][D][L]); 128-row tiles never
//                straddle a batch since 128 | 8192.
// WMMA f32 layouts per ISA 7.12.2:
//   A 16x4 : lane(0..15)/(16..31) -> M=lane&15, VGPR{0,1}=K{0,1}/{2,3}
//   B 4x16 : lane -> N=lane&15, VGPR{0,1}=K{0,1} (low half) / K{2,3} (high)
//   C 16x16: VGPR r -> M = r + 8*(lane>>4), N = lane&15
// ---------------------------------------------------------------------------
template <bool ATRANS>
__global__ void gemm_bias_wmma_kernel(const float* __restrict__ A,
                                      const float* __restrict__ W,
                                      const float* __restrict__ bias,
                                      float* __restrict__ C,
                                      int N) {
    constexpr int KTOT = 256;
    constexpr int KLD  = 36;             // k-stride (floats) for both tiles
    __shared__ float As[128 * KLD];      // [m][k]
    __shared__ float Bs[64 * KLD];       // [n][k]

    const int tid = threadIdx.x;
    const int wv  = tid >> 5;            // wave 0..7
    const int ln  = tid & 31;
    const int lh  = ln >> 4;             // half-wave -> K offset 2
    const int l15 = ln & 15;
    const int m0  = blockIdx.y * 128;
    const int n0  = blockIdx.x * 64;

    v8f zero = {0, 0, 0, 0, 0, 0, 0, 0};
    v8f acc0 = zero, acc1 = zero, acc2 = zero, acc3 = zero;

    const int batch = m0 / SEQ_LEN;      // only used when ATRANS
    const int lbase = m0 - batch * SEQ_LEN;

    for (int k0 = 0; k0 < KTOT; k0 += 32) {
        __syncthreads();
        if (!ATRANS) {
            int ml = tid >> 3;                 // 0..31
            int kq = (tid & 7) << 2;           // 0..28
            #pragma unroll
            for (int i = 0; i < 4; ++i, ml += 32) {
                const float4 va = *reinterpret_cast<const float4*>(
                    A + (size_t)(m0 + ml) * KTOT + k0 + kq);
                *reinterpret_cast<float4*>(&As[ml * KLD + kq]) = va;
            }
        } else {
            int kl = tid >> 5;                 // 0..7
            int mq = ln << 2;                  // 0..124
            #pragma unroll
            for (int i = 0; i < 4; ++i, kl += 8) {
                const float4 va = *reinterpret_cast<const float4*>(
                    A + ((size_t)(batch * KTOT + k0 + kl)) * SEQ_LEN + lbase + mq);
                As[(mq + 0) * KLD + kl] = va.x;    // transpose scatter
                As[(mq + 1) * KLD + kl] = va.y;
                As[(mq + 2) * KLD + kl] = va.z;
                As[(mq + 3) * KLD + kl] = va.w;
            }
        }
        {
            int nl = tid >> 3;                 // 0..31
            int kq = (tid & 7) << 2;
            #pragma unroll
            for (int i = 0; i < 2; ++i, nl += 32) {
                const float4 vb = *reinterpret_cast<const float4*>(
                    W + (size_t)(n0 + nl) * KTOT + k0 + kq);
                *reinterpret_cast<float4*>(&Bs[nl * KLD + kq]) = vb;
            }
        }
        __syncthreads();

        const int mrow = wv * 16 + l15;
        #pragma unroll
        for (int kk = 0; kk < 32; kk += 4) {
            const int kf = kk + lh * 2;        // even -> 8B aligned b64 loads
            const float2 a2 = *reinterpret_cast<const float2*>(&As[mrow * KLD + kf]);
            const float2 b0 = *reinterpret_cast<const float2*>(&Bs[(l15 +  0) * KLD + kf]);
            const float2 b1 = *reinterpret_cast<const float2*>(&Bs[(l15 + 16) * KLD + kf]);
            const float2 b2 = *reinterpret_cast<const float2*>(&Bs[(l15 + 32) * KLD + kf]);
            const float2 b3 = *reinterpret_cast<const float2*>(&Bs[(l15 + 48) * KLD + kf]);
            v2f af;  af.x  = a2.x; af.y  = a2.y;
            v2f bf0; bf0.x = b0.x; bf0.y = b0.y;
            v2f bf1; bf1.x = b1.x; bf1.y = b1.y;
            v2f bf2; bf2.x = b2.x; bf2.y = b2.y;
            v2f bf3; bf3.x = b3.x; bf3.y = b3.y;
            acc0 = __builtin_amdgcn_wmma_f32_16x16x4_f32(false, af, false, bf0, (short)0, acc0, false, false);
            acc1 = __builtin_amdgcn_wmma_f32_16x16x4_f32(false, af, false, bf1, (short)0, acc1, false, false);
            acc2 = __builtin_amdgcn_wmma_f32_16x16x4_f32(false, af, false, bf2, (short)0, acc2, false, false);
            acc3 = __builtin_amdgcn_wmma_f32_16x16x4_f32(false, af, false, bf3, (short)0, acc3, false, false);
        }
    }

    const int mbase = m0 + wv * 16 + lh * 8;
    v8f accs[4] = {acc0, acc1, acc2, acc3};
    #pragma unroll
    for (int j = 0; j < 4; ++j) {
        const int n  = n0 + j * 16 + l15;
        const float bv = bias[n];
        #pragma unroll
        for (int r = 0; r < 8; ++r)
            C[(size_t)(mbase + r) * N + n] = accs[j][r] + bv;
    }
}

// ---------------------------------------------------------------------------
// K2: split proj[B,L,768] into v/g0/g1, depthwise K=3 'same' conv on g0,g1,
// write all three transposed to [B,D,L] (coalesced reads and writes via LDS).
// Block handles a 64(l) x 64(d) tile with a +/-1 halo on l.
// ---------------------------------------------------------------------------
__global__ void transpose_dwconv_kernel(const float* __restrict__ proj,
                                        const float* __restrict__ c0W,
                                        const float* __restrict__ c0b,
                                        const float* __restrict__ c1W,
                                        const float* __restrict__ c1b,
                                        float* __restrict__ vT,
                                        float* __restrict__ g0c,
                                        float* __restrict__ g1c) {
    __shared__ float Ls[66][65];
    const int tid = threadIdx.x;
    const int l0  = blockIdx.x * 64;
    const int d0  = blockIdx.y * 64;
    const int b   = blockIdx.z;
    const int col = tid & 63;      // d within tile for loads
    const int rb  = tid >> 6;      // 0..3
    const int lll = tid & 63;      // l within tile for compute/writes

    for (int sec = 0; sec < 3; ++sec) {
        __syncthreads();
        for (int r = rb; r < 66; r += 4) {
            int gl = l0 - 1 + r;
            float v = 0.0f;
            if (gl >= 0 && gl < SEQ_LEN)
                v = proj[((size_t)(b * SEQ_LEN + gl)) * PROJ3 + sec * D_MODEL + d0 + col];
            Ls[r][col] = v;
        }
        __syncthreads();
        const float* cw = (sec == 1) ? c0W : c1W;
        const float* cb = (sec == 1) ? c0b : c1b;
        float* dstbuf   = (sec == 0) ? vT : ((sec == 1) ? g0c : g1c);
        for (int dl = rb; dl < 64; dl += 4) {
            const int d = d0 + dl;
            float outv;
            if (sec == 0) {
                outv = Ls[lll + 1][dl];
            } else {
                float w0 = cw[d * 3 + 0], w1 = cw[d * 3 + 1], w2 = cw[d * 3 + 2];
                outv = fmaf(w0, Ls[lll][dl],
                       fmaf(w1, Ls[lll + 1][dl],
                       fmaf(w2, Ls[lll + 2][dl], cb[d])));
            }
            dstbuf[((size_t)(b * D_MODEL + d)) * SEQ_LEN + l0 + lll] = outv;
        }
    }
}

// ---------------------------------------------------------------------------
// K3: per (b,d) row: FFT(v pad 2L) * Hf[d] -> IFFT -> *1/N -> * g0 * g1,
// overwrite the v row in place (safe: row fully loaded to LDS first).
// One workgroup per row. Dynamic LDS = 256KB FFT ping-pong + 32KB TDM staging
// (288KB total, legal on CDNA5's 320KB/WGP).
// The 32KB contiguous v row is fetched by the Tensor Data Mover:
// wave 0 issues tensor_load_to_lds with an ISA 8.3/8.4 descriptor
// (count=1, type=2, data_size=4B, tensor 8192x1, tile 8192x1, lds_addr=256KB),
// waits on TENSORcnt, then the workgroup expands it into the complex buffer.
// ---------------------------------------------------------------------------
#define FFT_LDS_BYTES   (2u * NFFT * 8u)        // 262144
#define STAGE_LDS_BYTES (SEQ_LEN * 4u)          // 32768

__global__ void fftconv_gate_kernel(float* __restrict__ vT,
                                    const float2* __restrict__ Hf,
                                    const float* __restrict__ g0c,
                                    const float* __restrict__ g1c) {
    extern __shared__ char smem[];
    float2* bufa = (float2*)smem;
    float2* bufb = bufa + NFFT;
    float*  stage = (float*)(smem + FFT_LDS_BYTES);
    const int tid = threadIdx.x;
    const int row = blockIdx.x;               // b*D + d
    const int d   = row & (D_MODEL - 1);
    float* vrow = vT + (size_t)row * SEQ_LEN;

#if __has_builtin(__builtin_amdgcn_tensor_load_to_lds)
    if (tid < 32) {                            // one TDM issue per workgroup
        const unsigned long long ga = (unsigned long long)vrow;
        v4u g0;                                // D# group 0 (ISA 8.3)
        g0.x = 1u;                             // count=1, user descriptor
        g0.y = FFT_LDS_BYTES;                  // lds_addr (bytes)
        g0.z = (unsigned)(ga & 0xFFFFFFFFull); // global_addr[31:0]
        g0.w = (unsigned)((ga >> 32) & 0x01FFFFFFull) | (2u << 30); // addr hi | type=2
        v8i g1;                                // D# group 1 (ISA 8.4)
        g1[0] = (int)(2u << 16);               // workgroup_mask=0, data_size=4B
        g1[1] = (int)((SEQ_LEN & 0xFFFF) << 16);       // tensor_dim0 lo16
        g1[2] = (int)((SEQ_LEN >> 16) | (1u << 16));   // tensor_dim0 hi | tensor_dim1=1
        g1[3] = (int)((unsigned)SEQ_LEN << 16);        // tile_dim0 = 8192
        g1[4] = 1;                             // tile_dim1=1, tile_dim2=0
        g1[5] = SEQ_LEN;                       // tensor_dim0_stride lo32
        g1[6] = 0;
        g1[7] = 0;
        v4i gz4 = {0, 0, 0, 0};
        v8i gz8 = {0, 0, 0, 0, 0, 0, 0, 0};
        __builtin_amdgcn_tensor_load_to_lds(g0, g1, gz4, gz4, gz8, 0);
        __builtin_amdgcn_s_wait_tensorcnt(0);
    }
    __syncthreads();
    for (int l = tid; l < NFFT; l += 256)
        bufa[l] = make_float2(l < SEQ_LEN ? stage[l] : 0.0f, 0.0f);
#else
    (void)stage;
    for (int l = tid; l < NFFT; l += 256)
        bufa[l] = make_float2(l < SEQ_LEN ? vrow[l] : 0.0f, 0.0f);
#endif
    __syncthreads();

    float2* spec = fft16384(bufa, bufb, tid, -1.0f);
    const float2* Hrow = Hf + (size_t)d * NFFT;
    for (int k = tid; k < NFFT; k += 256)
        spec[k] = cmulc(spec[k], Hrow[k]);
    __syncthreads();

    float2* other = (spec == bufa) ? bufb : bufa;
    float2* out   = fft16384(spec, other, tid, 1.0f);

    const float inv = 1.0f / (float)NFFT;
    const float* g0r = g0c + (size_t)row * SEQ_LEN;
    const float* g1r = g1c + (size_t)row * SEQ_LEN;
    for (int l = tid; l < SEQ_LEN; l += 256) {
        float yv = out[l].x * inv;
        vrow[l] = yv * g0r[l] * g1r[l];
    }
}

// ---------------------------------------------------------------------------
// Host-side launch. Workspace layout (bytes):
//   proj : 65536*768*4      = 201326592   @ 0
//   vT   : 2048*8192*4      =  67108864   @ 201326592   (reused as gated y)
//   g0c  :                     67108864   @ 268435456
//   g1c  :                     67108864   @ 335544320
//   Hf   : 256*16384*8      =  33554432   @ 402653184   (total 436207616)
// ---------------------------------------------------------------------------
extern "C" void kernel_launch(void* const* d_in, const int* in_sizes, int n_in,
                              void* d_out, int out_size, void* d_ws, size_t ws_size,
                              hipStream_t stream) {
    const float* x     = (const float*)d_in[0];
    const float* in_W  = (const float*)d_in[1];
    const float* in_b  = (const float*)d_in[2];
    const float* c0W   = (const float*)d_in[3];
    const float* c0b   = (const float*)d_in[4];
    const float* c1W   = (const float*)d_in[5];
    const float* c1b   = (const float*)d_in[6];
    const float* mW1   = (const float*)d_in[7];
    const float* mb1   = (const float*)d_in[8];
    const float* mW2   = (const float*)d_in[9];
    const float* mb2   = (const float*)d_in[10];
    const float* out_W = (const float*)d_in[11];
    const float* out_b = (const float*)d_in[12];
    const float* t     = (const float*)d_in[13];
    float* out = (float*)d_out;
    (void)in_sizes; (void)n_in; (void)out_size; (void)ws_size;

    char* ws = (char*)d_ws;
    float*  proj = (float*) (ws);
    float*  vT   = (float*) (ws + 201326592ull);
    float*  g0c  = (float*) (ws + 268435456ull);
    float*  g1c  = (float*) (ws + 335544320ull);
    float2* Hf   = (float2*)(ws + 402653184ull);

    const size_t fft_lds  = (size_t)FFT_LDS_BYTES;                      // 256 KB
    const size_t fftg_lds = (size_t)(FFT_LDS_BYTES + STAGE_LDS_BYTES);  // 288 KB

    hipLaunchKernelGGL(hfilter_fft_kernel, dim3(D_MODEL), dim3(256), fft_lds, stream,
                       t, mW1, mb1, mW2, mb2, Hf);
    hipLaunchKernelGGL((gemm_bias_wmma_kernel<false>),
                       dim3(PROJ3 / 64, (BATCH * SEQ_LEN) / 128), dim3(256), 0, stream,
                       x, in_W, in_b, proj, PROJ3);
    hipLaunchKernelGGL(transpose_dwconv_kernel,
                       dim3(SEQ_LEN / 64, D_MODEL / 64, BATCH), dim3(256), 0, stream,
                       proj, c0W, c0b, c1W, c1b, vT, g0c, g1c);
    hipLaunchKernelGGL(fftconv_gate_kernel, dim3(BATCH * D_MODEL), dim3(256), fftg_lds, stream,
                       vT, Hf, g0c, g1c);
    hipLaunchKernelGGL((gemm_bias_wmma_kernel<true>),
                       dim3(D_MODEL / 64, (BATCH * SEQ_LEN) / 128), dim3(256), 0, stream,
                       vT, out_W, out_b, out, D_MODEL);
}